// inverse_retina_polar_batch_att_slow_89017492177078
// MI455X (gfx1250) — compile-verified
//
#include <hip/hip_runtime.h>
#include <math.h>

typedef __attribute__((ext_vector_type(2))) float v2f;
typedef __attribute__((ext_vector_type(8))) float v8f;

#define HIN 224
#define WIN 224
#define B_ 8
#define C_ 64
#define PLANE (HIN * WIN)            // 50176
#define GH 64
#define GW 128
#define GPTS (GH * GW)               // 8192
#define OUT_PLANE 8192               // 64*128
#define POOLED_ELEMS (B_ * C_ * OUT_PLANE)

// ws layout (in floats)
#define WS_BRANCH 0                  // 512 floats
#define WS_WEIGHT 512                // 16 floats
#define WS_GRID   1024               // 8*8192*2 floats

// ---------------- Stage 1: global average pool per (b,c) ----------------
__global__ void k_mean(const float* __restrict__ x, float* __restrict__ ws) {
  int bc = blockIdx.x;            // 0..511
  int tid = threadIdx.x;          // 0..255
  const float4* p = (const float4*)(x + (size_t)bc * PLANE);
  float s = 0.f;
  // 50176/4 = 12544 float4 = 256 * 49
  for (int i = tid; i < PLANE / 4; i += 256) {
    float4 v = p[i];
    s += v.x + v.y + v.z + v.w;
  }
  __shared__ float red[256];
  red[tid] = s;
  __syncthreads();
  for (int off = 128; off > 0; off >>= 1) {
    if (tid < off) red[tid] += red[tid + off];
    __syncthreads();
  }
  if (tid == 0) ws[WS_BRANCH + bc] = red[0] / (float)PLANE;
}

// ---------------- Stage 2: tiny MLP via f32 WMMA (exact precision) ------
// Layer1: [16(pad of 8) x 64] @ [64 x 32] via V_WMMA_F32_16X16X4_F32,
//         K in 16 steps of 4, N in two 16-wide tiles.
__global__ void k_mlp(const float* __restrict__ w1, const float* __restrict__ b1,
                      const float* __restrict__ w2, const float* __restrict__ b2,
                      float* __restrict__ ws, float* __restrict__ out_tail) {
  const int lane = threadIdx.x;       // 0..31 (single wave32, EXEC all 1s)
  const int half = lane >> 4;         // 0: K pair (k0,k0+1), 1: (k0+2,k0+3)
  const int m    = lane & 15;         // A row (lanes 0-15) / B col within tile
  const float* branch = ws + WS_BRANCH;

  v8f acc0 = {};
  v8f acc1 = {};
  for (int kk = 0; kk < 16; ++kk) {
    int k = kk * 4 + 2 * half;
    // A 16x4 f32 layout: lane L<16: V0=A[m][k0], V1=A[m][k0+1];
    //                    lane L>=16: V0=A[m][k0+2], V1=A[m][k0+3]
    v2f a;
    a.x = (m < 8) ? branch[m * 64 + k]     : 0.f;   // rows 8..15 are zero pad
    a.y = (m < 8) ? branch[m * 64 + k + 1] : 0.f;
    // B 4x16 f32: row striped across lanes; rows split across lane halves
    v2f bb0, bb1;
    bb0.x = w1[(k)     * 32 + m];
    bb0.y = w1[(k + 1) * 32 + m];
    bb1.x = w1[(k)     * 32 + 16 + m];
    bb1.y = w1[(k + 1) * 32 + 16 + m];
    acc0 = __builtin_amdgcn_wmma_f32_16x16x4_f32(false, a, false, bb0,
                                                 (short)0, acc0, false, false);
    acc1 = __builtin_amdgcn_wmma_f32_16x16x4_f32(false, a, false, bb1,
                                                 (short)0, acc1, false, false);
  }

  // C/D layout: VGPR v, lanes 0-15 -> M=v, N=lane; lanes 16-31 -> M=v+8.
  // We need M=0..7 only -> lanes 0..15, VGPRs 0..7.
  __shared__ float h[8][32];
  if (half == 0) {
    for (int v = 0; v < 8; ++v) {
      float h0 = acc0[v] + b1[m];
      float h1 = acc1[v] + b1[16 + m];
      h[v][m]      = h0 > 0.f ? h0 : 0.f;
      h[v][16 + m] = h1 > 0.f ? h1 : 0.f;
    }
  }
  __syncthreads();

  // Layer 2: [8x32] @ [32x2] + b2, sigmoid. 16 lanes, one output each.
  if (lane < 16) {
    int mm = lane >> 1, j = lane & 1;
    float s = b2[j];
    for (int k = 0; k < 32; ++k) s += h[mm][k] * w2[k * 2 + j];
    float wgt = 1.f / (1.f + expf(-s));
    ws[WS_WEIGHT + mm * 2 + j] = wgt;
    out_tail[mm * 2 + j] = wgt;        // second tuple element of the output
  }
}

// ---------------- Stage 3: log-polar grid (top-left 64x128 corner) ------
__global__ void k_grid(const float* __restrict__ l_t, float* __restrict__ ws) {
  int idx = blockIdx.x * 256 + threadIdx.x;   // 0..65535 = 8*64*128
  int b   = idx >> 13;
  int rem = idx & 8191;
  int i   = rem >> 7;      // 0..63  (H dim)
  int j   = rem & 127;     // 0..127 (W dim)

  float w0  = ws[WS_WEIGHT + b * 2 + 0];
  float w1v = ws[WS_WEIGHT + b * 2 + 1];
  float lo = logf(w0 * 0.01f);     // log(weight0 * R_MIN)
  float hi = logf(w1v * 0.6f);     // log(weight1 * R_MAX)

  float xg = (float)(i - 32) * (1.f / 32.f);
  float yg = (float)(j - 64) * (1.f / 64.f);
  float logr = logf(fmaxf(sqrtf(xg * xg + yg * yg), 1e-12f));
  float r = 64.f * (logr - lo) / (hi - lo);
  float a = atan2f(yg, xg);
  if (!(a > 0.f)) a = 6.28318530717958647692f + a;
  float gx = a * (1.f / 3.14159265358979323846f) - 1.f;  // t/(RW/2)-1 == a/pi-1
  float gy = r * (1.f / 32.f) - 1.f;                      // r/(RH/2)-1

  float2* g = (float2*)(ws + WS_GRID);
  g[idx] = make_float2(gx + l_t[b * 2 + 0], gy + l_t[b * 2 + 1]);
}

// ---------------- Stage 4: bilinear sample + 4x4 pool -------------------
__device__ __forceinline__ float bilin(const float* __restrict__ xp,
                                       float gx, float gy) {
  float ix = ((gx + 1.f) * 224.f - 1.f) * 0.5f;
  float iy = ((gy + 1.f) * 224.f - 1.f) * 0.5f;
  ix = fminf(fmaxf(ix, 0.f), 223.f);
  iy = fminf(fmaxf(iy, 0.f), 223.f);
  float x0f = floorf(ix), y0f = floorf(iy);
  float wx = ix - x0f, wy = iy - y0f;
  int x0 = (int)x0f, y0 = (int)y0f;
  int x1 = x0 + 1; if (x1 > 223) x1 = 223;
  int y1 = y0 + 1; if (y1 > 223) y1 = 223;
  const float* r0 = xp + y0 * 224;
  const float* r1 = xp + y1 * 224;
  float v00 = r0[x0], v01 = r0[x1], v10 = r1[x0], v11 = r1[x1];
  return v00 * (1.f - wx) * (1.f - wy) + v01 * wx * (1.f - wy)
       + v10 * (1.f - wx) * wy         + v11 * wx * wy;
}

__global__ void k_sample(const float* __restrict__ x, const float* __restrict__ l_t,
                         const float* __restrict__ ws, float* __restrict__ out) {
  int bc = blockIdx.x;                 // 0..511
  int b  = bc >> 6;
  int tid = threadIdx.x;               // 0..255
  const float* xp = x + (size_t)bc * PLANE;
  const float2* g = (const float2*)(ws + WS_GRID) + (size_t)b * GPTS;

  // Corner outputs (h<16, w<32): 16-sample averages -> LDS
  __shared__ float corner[512];
  for (int o = tid; o < 512; o += 256) {
    int h = o >> 5, w = o & 31;
    float acc = 0.f;
    for (int p = 0; p < 4; ++p) {
      const float2* row = g + (4 * h + p) * 128 + 4 * w;
      for (int q = 0; q < 4; ++q) {
        float2 gv = row[q];
        acc += bilin(xp, gv.x, gv.y);
      }
    }
    corner[o] = acc * (1.f / 16.f);
  }
  // Everything outside the corner pools 16 identical samples at grid=(l_t)
  float sc = bilin(xp, l_t[b * 2 + 0], l_t[b * 2 + 1]);
  __syncthreads();

  float* op = out + (size_t)bc * OUT_PLANE;
  for (int o = tid; o < OUT_PLANE; o += 256) {
    int h = o >> 7, w = o & 127;
    op[o] = (h < 16 && w < 32) ? corner[(h << 5) | w] : sc;
  }
}

extern "C" void kernel_launch(void* const* d_in, const int* in_sizes, int n_in,
                              void* d_out, int out_size, void* d_ws, size_t ws_size,
                              hipStream_t stream) {
  const float* x   = (const float*)d_in[0];
  const float* l_t = (const float*)d_in[1];
  const float* w1  = (const float*)d_in[2];
  const float* b1  = (const float*)d_in[3];
  const float* w2  = (const float*)d_in[4];
  const float* b2  = (const float*)d_in[5];
  float* out = (float*)d_out;
  float* ws  = (float*)d_ws;   // needs ~516 KB

  k_mean  <<<B_ * C_, 256, 0, stream>>>(x, ws);
  k_mlp   <<<1, 32, 0, stream>>>(w1, b1, w2, b2, ws, out + POOLED_ELEMS);
  k_grid  <<<(B_ * GPTS) / 256, 256, 0, stream>>>(l_t, ws);
  k_sample<<<B_ * C_, 256, 0, stream>>>(x, l_t, ws, out);
}